// GAT_10282151706869
// MI455X (gfx1250) — compile-verified
//
#include <hip/hip_runtime.h>
#include <hip/hip_bf16.h>
#include <math.h>

// ---------------------------------------------------------------------------
// GAT (2-layer, heads=1) for MI455X / gfx1250, wave32.
// Dense GEMMs: V_WMMA_F32_16X16X4_F32 (fp32 matrix cores; inputs are fp32).
// Sparse edge phase: fp32 global atomics (h/agg/edge lists fit in 192MB L2,
// so the scatter is L2-bound; 23.3 TB/s HBM is never the limiter).
// x (102MB, read once) is loaded non-temporally to preserve L2 residency of
// the reused h1/agg1 buffers.
// ---------------------------------------------------------------------------

typedef __attribute__((ext_vector_type(2))) float v2f;
typedef __attribute__((ext_vector_type(4))) float v4f;
typedef __attribute__((ext_vector_type(8))) float v8f;

// -------------------- small utility kernels --------------------

__global__ void fill_f32_kernel(float* __restrict__ p, float v, long long n) {
    long long i = (long long)blockIdx.x * blockDim.x + threadIdx.x;
    if (i < n) p[i] = v;
}

// float atomic max via monotone integer mapping (mixed signs OK,
// buffer initialized to -inf = 0xFF800000).
__device__ inline void atomicMaxF32(float* addr, float val) {
    if (!signbit(val)) {
        atomicMax((int*)addr, __float_as_int(val));
    } else {
        atomicMin((unsigned int*)addr, (unsigned int)__float_as_int(val));
    }
}

// -------------------- WMMA GEMM 1: h1 = x @ W1 --------------------
// One block = one 16-row M tile staged in LDS (non-temporal A loads);
// 8 waves each own one 16-col N tile (N == 128); K-loop in steps of 4
// feeding v_wmma_f32_16x16x4_f32.
//
// Fragment layouts per ISA 7.12.2 (wave32):
//  A 16x4 : lanes 0-15 -> {K=0,K=1}, lanes 16-31 -> {K=2,K=3}, row M = lane&15
//  B 4x16 : mirrored (lanes 0-15 rows K=0/1, lanes 16-31 rows K=2/3, col = lane&15)
//  C/D    : VGPR r: lanes 0-15 row M=r, lanes 16-31 row M=r+8, col N = lane&15
template <int KDIM>
__global__ __launch_bounds__(256) void gemm1_wmma_kernel(
    const float* __restrict__ A, const float* __restrict__ B,
    float* __restrict__ D, int M, int N) {
    __shared__ float As[16 * KDIM];

    const int mt   = blockIdx.x;
    const int row0 = mt * 16;
    const int tid  = threadIdx.x;

    // cooperative load of the 16 x KDIM A tile (b128, non-temporal,
    // rows past M zero-filled)
    const int nvec = (16 * KDIM) / 4;
    for (int v = tid; v < nvec; v += 256) {
        int r  = v / (KDIM / 4);
        int c4 = v % (KDIM / 4);
        v4f val = {0.f, 0.f, 0.f, 0.f};
        int gr = row0 + r;
        if (gr < M)
            val = __builtin_nontemporal_load(
                      (const v4f*)(A + (size_t)gr * KDIM + c4 * 4));
        *(v4f*)(As + r * KDIM + c4 * 4) = val;
    }
    __syncthreads();

    const int wave = tid >> 5;        // N tile id (0..7)
    const int lane = tid & 31;
    const int col0 = wave * 16;
    const int idx  = lane & 15;       // row for A frag, col for B frag
    const int kk   = (lane >> 4) * 2; // K sub-offset for this lane half
    const int gcB  = min(col0 + idx, N - 1);   // clamped, branch-free loads

    v8f c = {};
    for (int k = 0; k < KDIM; k += 4) {
        v2f a, b;
        v2f av = *(const v2f*)(As + idx * KDIM + k + kk);   // ds_load_b64
        a.x = av.x; a.y = av.y;
        b.x = B[(size_t)(k + kk) * N + gcB];
        b.y = B[(size_t)(k + kk + 1) * N + gcB];
        c = __builtin_amdgcn_wmma_f32_16x16x4_f32(
                /*neg_a=*/false, a, /*neg_b=*/false, b,
                /*c_mod=*/(short)0, c, /*reuse_a=*/false, /*reuse_b=*/false);
    }

    const int half = lane >> 4;
    const int gc   = col0 + idx;
    for (int r = 0; r < 8; ++r) {
        int gr = row0 + r + 8 * half;
        if (gr < M && gc < N) D[(size_t)gr * N + gc] = c[r];
    }
}

// -------------------- WMMA GEMM 2: h2 = relu(agg1 + b1) @ W2 --------------------
// Generic M,K,N; one wave per 16x16 output tile; relu(+bias) fused on A loads.
// Inner loop is branch-free: out-of-range rows/cols are clamped on load (they
// only feed output elements that are never stored).
__global__ __launch_bounds__(256) void gemm2_wmma_kernel(
    const float* __restrict__ A, const float* __restrict__ abias,
    const float* __restrict__ B, float* __restrict__ D,
    int M, int K, int N) {
    const long long gtid = (long long)blockIdx.x * blockDim.x + threadIdx.x;
    const long long wid  = gtid >> 5;
    const int lane = threadIdx.x & 31;

    const int numNt = (N + 15) / 16;
    const int numMt = (M + 15) / 16;
    if (wid >= (long long)numMt * numNt) return;   // whole wave exits together

    const int mt = (int)(wid / numNt);
    const int nt = (int)(wid % numNt);
    const int row0 = mt * 16, col0 = nt * 16;
    const int idx  = lane & 15;
    const int half = lane >> 4;
    const int kk   = half * 2;

    const int grA = min(row0 + idx, M - 1);  // clamped A row
    const int gcB = min(col0 + idx, N - 1);  // clamped B col
    const float* __restrict__ Arow = A + (size_t)grA * K;

    v8f c = {};
    for (int k = 0; k < K; k += 4) {
        v2f a, b;
        v2f av = *(const v2f*)(Arow + k + kk);          // global_load_b64
        v2f bb = *(const v2f*)(abias + k + kk);
        a.x = fmaxf(av.x + bb.x, 0.f);
        a.y = fmaxf(av.y + bb.y, 0.f);
        b.x = B[(size_t)(k + kk) * N + gcB];
        b.y = B[(size_t)(k + kk + 1) * N + gcB];
        c = __builtin_amdgcn_wmma_f32_16x16x4_f32(
                false, a, false, b, (short)0, c, false, false);
    }

    const int gc = col0 + idx;
    for (int r = 0; r < 8; ++r) {
        int orow = row0 + r + 8 * half;
        if (orow < M && gc < N) D[(size_t)orow * N + gc] = c[r];
    }
}

// -------------------- attention score dot products (wave per node) ------------
__global__ __launch_bounds__(256) void scores_kernel(
    const float* __restrict__ h, const float* __restrict__ a_src,
    const float* __restrict__ a_dst, float* __restrict__ s_src,
    float* __restrict__ s_dst, int N, int F) {
    long long wid = ((long long)blockIdx.x * blockDim.x + threadIdx.x) >> 5;
    int lane = threadIdx.x & 31;
    if (wid >= N) return;
    const float* row = h + (size_t)wid * F;
    float ss = 0.f, sd = 0.f;
    for (int f = lane; f < F; f += 32) {      // coalesced b32 loads
        float v = row[f];
        ss += v * a_src[f];
        sd += v * a_dst[f];
    }
    for (int o = 16; o > 0; o >>= 1) {        // wave32 butterfly reduction
        ss += __shfl_xor(ss, o, 32);
        sd += __shfl_xor(sd, o, 32);
    }
    if (lane == 0) {
        s_src[wid] = ss;
        s_dst[wid] = sd;
    }
}

// -------------------- edge phase --------------------
// Edge i in [0, E): src = ei[i], dst = ei[E+i]; i in [E, E+N): self loop.
__device__ inline void edge_decode(const long long* __restrict__ ei,
                                   long long E, long long i, int& s, int& d) {
    if (i < E) { s = (int)ei[i]; d = (int)ei[E + i]; }
    else       { s = d = (int)(i - E); }
}

__global__ void edge_e_kernel(const long long* __restrict__ ei, long long E, int Nn,
                              const float* __restrict__ s_src,
                              const float* __restrict__ s_dst,
                              float* __restrict__ ebuf, float* __restrict__ mbuf) {
    long long i = (long long)blockIdx.x * blockDim.x + threadIdx.x;
    long long total = E + Nn;
    if (i >= total) return;
    int s, d;
    edge_decode(ei, E, i, s, d);
    float e = s_src[s] + s_dst[d];
    e = e > 0.f ? e : 0.2f * e;   // leaky_relu, NEG_SLOPE = 0.2
    ebuf[i] = e;
    atomicMaxF32(&mbuf[d], e);
}

__global__ void edge_exp_kernel(const long long* __restrict__ ei, long long E, int Nn,
                                const float* __restrict__ mbuf,
                                float* __restrict__ ebuf,   // in: e, out: exp(e-m)
                                float* __restrict__ dbuf) {
    long long i = (long long)blockIdx.x * blockDim.x + threadIdx.x;
    long long total = E + Nn;
    if (i >= total) return;
    int s, d;
    edge_decode(ei, E, i, s, d);
    float ex = __expf(ebuf[i] - mbuf[d]);
    ebuf[i] = ex;
    atomicAdd(&dbuf[d], ex);
}

// one wave per edge: agg[dst] += (ex/denom[dst]) * h[src]
__global__ __launch_bounds__(256) void edge_agg_kernel(
    const long long* __restrict__ ei, long long E, int Nn,
    const float* __restrict__ ebuf, const float* __restrict__ dbuf,
    const float* __restrict__ h, float* __restrict__ agg, int F) {
    long long gtid = (long long)blockIdx.x * blockDim.x + threadIdx.x;
    long long wid  = gtid >> 5;
    int lane = threadIdx.x & 31;
    long long total = E + Nn;
    if (wid >= total) return;
    int s, d;
    edge_decode(ei, E, wid, s, d);
    const float* hs = h + (size_t)s * F;
    float* ad = agg + (size_t)d * F;
    __builtin_prefetch(hs + lane, 0, 3);      // global_prefetch_b8 (gfx1250)
    float alpha = ebuf[wid] / dbuf[d];
    for (int f = lane; f < F; f += 32)
        atomicAdd(&ad[f], alpha * hs[f]);     // L2-resident fp32 atomics
}

// -------------------- fused (+b2) log_softmax (wave per node, in place) -------
__global__ __launch_bounds__(256) void logsoftmax_bias_kernel(
    float* __restrict__ out, const float* __restrict__ bias, int N, int C) {
    long long wid = ((long long)blockIdx.x * blockDim.x + threadIdx.x) >> 5;
    int lane = threadIdx.x & 31;
    if (wid >= N) return;
    float* row = out + (size_t)wid * C;
    float mx = -INFINITY;
    for (int j = lane; j < C; j += 32) mx = fmaxf(mx, row[j] + bias[j]);
    for (int o = 16; o > 0; o >>= 1) mx = fmaxf(mx, __shfl_xor(mx, o, 32));
    float sum = 0.f;
    for (int j = lane; j < C; j += 32) sum += __expf(row[j] + bias[j] - mx);
    for (int o = 16; o > 0; o >>= 1) sum += __shfl_xor(sum, o, 32);
    float lse = mx + __logf(sum);
    for (int j = lane; j < C; j += 32) row[j] = row[j] + bias[j] - lse;
}

// ---------------------------------------------------------------------------
// host-side launch
// ---------------------------------------------------------------------------
extern "C" void kernel_launch(void* const* d_in, const int* in_sizes, int n_in,
                              void* d_out, int out_size, void* d_ws, size_t ws_size,
                              hipStream_t stream) {
    // setup_inputs order: x, edge_index, W1, a_src1, a_dst1, b1, W2, a_src2, a_dst2, b2
    const float*     x      = (const float*)d_in[0];
    const long long* ei     = (const long long*)d_in[1];  // int64 per reference
    const float*     W1     = (const float*)d_in[2];
    const float*     a_src1 = (const float*)d_in[3];
    const float*     a_dst1 = (const float*)d_in[4];
    const float*     b1     = (const float*)d_in[5];
    const float*     W2     = (const float*)d_in[6];
    const float*     a_src2 = (const float*)d_in[7];
    const float*     a_dst2 = (const float*)d_in[8];
    const float*     b2     = (const float*)d_in[9];

    const int H    = in_sizes[3];            // 128
    const int C    = in_sizes[7];            // 40
    const int Fin  = in_sizes[2] / H;        // 256
    const int N    = in_sizes[0] / Fin;      // 100000
    const long long E = in_sizes[1] / 2;     // 1600000
    const long long tot = E + N;             // edges incl. self loops

    // ---- workspace layout (256B aligned slices) ----
    char* base = (char*)d_ws;
    size_t off = 0;
    auto take = [&](size_t bytes) -> char* {
        char* p = base + off;
        off = (off + bytes + 255) & ~(size_t)255;
        return p;
    };
    float* h1    = (float*)take((size_t)N * H * sizeof(float));
    float* agg1  = (float*)take((size_t)N * H * sizeof(float));
    float* h2    = (float*)take((size_t)N * C * sizeof(float));
    float* ssrc  = (float*)take((size_t)N * sizeof(float));
    float* sdst  = (float*)take((size_t)N * sizeof(float));
    float* mbuf  = (float*)take((size_t)N * sizeof(float));
    float* dbuf  = (float*)take((size_t)N * sizeof(float));
    float* ebuf  = (float*)take((size_t)tot * sizeof(float));
    float* out   = (float*)d_out;            // agg2 accumulated here, then in-place logsoftmax

    const int T = 256;
    auto blocks1d = [](long long n, int t) { return (unsigned)((n + t - 1) / t); };

    // ================= layer 1 =================
    gemm1_wmma_kernel<256><<<(N + 15) / 16, T, 0, stream>>>(x, W1, h1, N, H);
    scores_kernel<<<blocks1d((long long)N * 32, T), T, 0, stream>>>(
        h1, a_src1, a_dst1, ssrc, sdst, N, H);
    fill_f32_kernel<<<blocks1d(N, T), T, 0, stream>>>(mbuf, -INFINITY, N);
    fill_f32_kernel<<<blocks1d(N, T), T, 0, stream>>>(dbuf, 0.f, N);
    fill_f32_kernel<<<blocks1d((long long)N * H, T), T, 0, stream>>>(agg1, 0.f, (long long)N * H);
    edge_e_kernel  <<<blocks1d(tot, T), T, 0, stream>>>(ei, E, N, ssrc, sdst, ebuf, mbuf);
    edge_exp_kernel<<<blocks1d(tot, T), T, 0, stream>>>(ei, E, N, mbuf, ebuf, dbuf);
    edge_agg_kernel<<<blocks1d(tot * 32, T), T, 0, stream>>>(ei, E, N, ebuf, dbuf, h1, agg1, H);

    // ================= layer 2 =================
    {
        long long waves = (long long)((N + 15) / 16) * ((C + 15) / 16);
        gemm2_wmma_kernel<<<blocks1d(waves * 32, T), T, 0, stream>>>(
            agg1, b1, W2, h2, N, H, C);
    }
    scores_kernel<<<blocks1d((long long)N * 32, T), T, 0, stream>>>(
        h2, a_src2, a_dst2, ssrc, sdst, N, C);
    fill_f32_kernel<<<blocks1d(N, T), T, 0, stream>>>(mbuf, -INFINITY, N);
    fill_f32_kernel<<<blocks1d(N, T), T, 0, stream>>>(dbuf, 0.f, N);
    fill_f32_kernel<<<blocks1d((long long)N * C, T), T, 0, stream>>>(out, 0.f, (long long)N * C);
    edge_e_kernel  <<<blocks1d(tot, T), T, 0, stream>>>(ei, E, N, ssrc, sdst, ebuf, mbuf);
    edge_exp_kernel<<<blocks1d(tot, T), T, 0, stream>>>(ei, E, N, mbuf, ebuf, dbuf);
    edge_agg_kernel<<<blocks1d(tot * 32, T), T, 0, stream>>>(ei, E, N, ebuf, dbuf, h2, out, C);

    // out = log_softmax(out + b2) in place
    logsoftmax_bias_kernel<<<blocks1d((long long)N * 32, T), T, 0, stream>>>(out, b2, N, C);
}